// WNBAGNN_66829691126287
// MI455X (gfx1250) — compile-verified
//
#include <hip/hip_runtime.h>

// ---------------- problem constants (match reference) ----------------
#define N_NODES 20000
#define E_EDGES 320000
#define ET      (E_EDGES + N_NODES)   // edges + self loops = 340000
#define FEAT    64
#define H       128
#define OUT_F   10

typedef __attribute__((ext_vector_type(2))) float v2f;
typedef __attribute__((ext_vector_type(8))) float v8f;

__device__ __forceinline__ float lrelu(float v) { return v > 0.f ? v : 0.2f * v; }

__device__ __forceinline__ float wave_max(float v) {
#pragma unroll
    for (int o = 16; o > 0; o >>= 1) v = fmaxf(v, __shfl_xor(v, o, 32));
    return v;
}
__device__ __forceinline__ float wave_sum(float v) {
#pragma unroll
    for (int o = 16; o > 0; o >>= 1) v += __shfl_xor(v, o, 32);
    return v;
}

// ---------------- f32 WMMA 16x16 tile: A from LDS, B from global -------------
// A fragment (f32 16x4): lane l -> m = l&15, g = l>>4; VGPR v holds K = 2g+v.
// B fragment (f32 4x16): lane l -> n = l&15, g = l>>4; VGPR v holds K = 2g+v.
// C/D (f32 16x16):       lane l -> n = l&15, g = l>>4; VGPR v holds M = 8g+v.
// NOTE: branchless inner loop — caller guarantees all 16 B columns are valid.
__device__ __forceinline__ v8f wmma_tile_lds(const float* __restrict__ As /*16 x K*/,
                                             int K,
                                             const float* __restrict__ B, int ldb,
                                             int col0, int lane) {
    const int m = lane & 15, g = lane >> 4, n = lane & 15;
    const float* Bp = B + col0 + n + (size_t)(2 * g) * ldb;
    const float* Ap = As + m * K + 2 * g;
    v8f acc = {0.f, 0.f, 0.f, 0.f, 0.f, 0.f, 0.f, 0.f};
#pragma unroll 4
    for (int k = 0; k < K; k += 4) {
        v2f a, b;
        a.x = Ap[k + 0];
        a.y = Ap[k + 1];
        b.x = Bp[(size_t)k * ldb];
        b.y = Bp[(size_t)(k + 1) * ldb];
        acc = __builtin_amdgcn_wmma_f32_16x16x4_f32(false, a, false, b,
                                                    (short)0, acc, false, false);
    }
    return acc;
}

__device__ __forceinline__ void store_tile(float* __restrict__ C, int ldc,
                                           int row0, int col0,
                                           v8f acc, const float* __restrict__ bias,
                                           bool relu, int lane) {
    const int g = lane >> 4, n = lane & 15;
    const float bv = bias ? bias[col0 + n] : 0.f;
#pragma unroll
    for (int v = 0; v < 8; ++v) {
        float val = acc[v] + bv;
        if (relu) val = fmaxf(val, 0.f);
        C[(size_t)(row0 + 8 * g + v) * ldc + col0 + n] = val;
    }
}

// ---------------- kernel 0: embed-gather + projection GEMM + ReLU ------------
// x[N,128] = relu( concat(emb[ids], feats)[N,192] @ proj_W[192,128] + proj_b )
__global__ void k_proj(const float* __restrict__ xp, const float* __restrict__ emb,
                       const float* __restrict__ W, const float* __restrict__ bias,
                       float* __restrict__ x) {
    __shared__ float As[16 * 192];
    const int row0 = blockIdx.x * 16;
    const int tid = threadIdx.x;
    for (int idx = tid; idx < 16 * 192; idx += blockDim.x) {
        const int r = row0 + idx / 192, k = idx % 192;
        float v;
        if (k < H) {
            const int id = (int)xp[(size_t)r * (FEAT + 1)];
            v = emb[(size_t)id * H + k];
        } else {
            v = xp[(size_t)r * (FEAT + 1) + 1 + (k - H)];
        }
        As[idx] = v;
    }
    __syncthreads();
    const int wave = tid >> 5, lane = tid & 31;
    const int col0 = (blockIdx.y * 4 + wave) * 16;   // 8 col tiles over y=0..1
    v8f acc = wmma_tile_lds(As, 192, W, H, col0, lane);
    store_tile(x, H, row0, col0, acc, bias, true, lane);
}

// ---------------- generic WMMA GEMM (ncols % 16 == 0, branchless) ------------
__global__ void k_gemm(const float* __restrict__ A, int lda,
                       const float* __restrict__ B, int ldb,
                       const float* __restrict__ bias,
                       float* __restrict__ C, int ldc,
                       int K, int relu) {
    extern __shared__ float As[];
    const int row0 = blockIdx.x * 16;
    const int tid = threadIdx.x;
    for (int idx = tid; idx < 16 * K; idx += blockDim.x)
        As[idx] = A[(size_t)(row0 + idx / K) * lda + (idx % K)];
    __syncthreads();
    const int wave = tid >> 5, lane = tid & 31;
    const int col0 = (blockIdx.y * 4 + wave) * 16;
    v8f acc = wmma_tile_lds(As, K, B, ldb, col0, lane);
    store_tile(C, ldc, row0, col0, acc, bias, relu != 0, lane);
}

// ---------------- output head: N=10 via zero-padded LDS B tile ---------------
// out[N,10] = x2[N,128] @ out_W[128,10] + out_b  (B padded to 128x16 in LDS)
__global__ void k_out(const float* __restrict__ A, const float* __restrict__ Bw,
                      const float* __restrict__ bias, float* __restrict__ C) {
    __shared__ float As[16 * H];
    __shared__ float Bs[H * 16];
    const int row0 = blockIdx.x * 16;
    const int tid = threadIdx.x;
    for (int idx = tid; idx < 16 * H; idx += blockDim.x)
        As[idx] = A[(size_t)(row0 + idx / H) * H + (idx % H)];
    for (int idx = tid; idx < H * 16; idx += blockDim.x) {
        const int r = idx / 16, c = idx % 16;
        Bs[idx] = (c < OUT_F) ? Bw[r * OUT_F + c] : 0.f;
    }
    __syncthreads();
    if (tid >= 32) return;                 // wave 0 computes; others only staged
    const int lane = tid;
    const int m = lane & 15, g = lane >> 4, n = lane & 15;
    v8f acc = {0.f, 0.f, 0.f, 0.f, 0.f, 0.f, 0.f, 0.f};
#pragma unroll
    for (int k = 0; k < H; k += 4) {       // fully unrolled: 32 straight WMMAs
        v2f a, b;
        a.x = As[m * H + k + 2 * g + 0];
        a.y = As[m * H + k + 2 * g + 1];
        b.x = Bs[(k + 2 * g + 0) * 16 + n];
        b.y = Bs[(k + 2 * g + 1) * 16 + n];
        acc = __builtin_amdgcn_wmma_f32_16x16x4_f32(false, a, false, b,
                                                    (short)0, acc, false, false);
    }
    if (n < OUT_F) {
        const float bv = bias[n];
#pragma unroll
        for (int v = 0; v < 8; ++v)
            C[(size_t)(row0 + 8 * g + v) * OUT_F + n] = acc[v] + bv;
    }
}

// ---------------- CSR-by-dst construction (no fp atomics downstream) ---------
__global__ void k_zero(int* __restrict__ counts) {
    const int i = blockIdx.x * blockDim.x + threadIdx.x;
    if (i < N_NODES) counts[i] = 0;
}

__global__ void k_count(const int* __restrict__ ei, int* __restrict__ counts) {
    const int i = blockIdx.x * blockDim.x + threadIdx.x;
    if (i >= ET) return;
    const int dst = (i < E_EDGES) ? ei[E_EDGES + i] : i - E_EDGES;
    atomicAdd(&counts[dst], 1);
}

// single-block exclusive scan (20000 elems, 1024 threads, Hillis-Steele chunks)
__global__ void k_scan(const int* __restrict__ counts, int* __restrict__ offs,
                       int* __restrict__ cursor) {
    __shared__ int buf[1024];
    __shared__ int carry;
    const int tid = threadIdx.x;
    if (tid == 0) carry = 0;
    __syncthreads();
    for (int base = 0; base < N_NODES; base += 1024) {
        const int v = (base + tid < N_NODES) ? counts[base + tid] : 0;
        buf[tid] = v;
        __syncthreads();
        for (int o = 1; o < 1024; o <<= 1) {
            const int t = (tid >= o) ? buf[tid - o] : 0;
            __syncthreads();
            buf[tid] += t;
            __syncthreads();
        }
        const int excl = buf[tid] - v;
        if (base + tid < N_NODES) {
            offs[base + tid] = carry + excl;
            cursor[base + tid] = carry + excl;
        }
        __syncthreads();
        if (tid == 1023) carry += buf[1023];
        __syncthreads();
    }
    if (tid == 0) offs[N_NODES] = carry; // == ET
}

__global__ void k_scatter(const int* __restrict__ ei, int* __restrict__ cursor,
                          int* __restrict__ eid) {
    const int i = blockIdx.x * blockDim.x + threadIdx.x;
    if (i >= ET) return;
    const int dst = (i < E_EDGES) ? ei[E_EDGES + i] : i - E_EDGES;
    const int pos = atomicAdd(&cursor[dst], 1);
    eid[pos] = i;
}

// ---------------- layer-1 edge scores: e1[E,4] --------------------------------
__global__ void k_scores1(const int* __restrict__ ei, const float* __restrict__ xl,
                          const float* __restrict__ xr, const float* __restrict__ att,
                          float* __restrict__ e1) {
    const int wave = threadIdx.x >> 5, lane = threadIdx.x & 31;
    const int edge = blockIdx.x * 8 + wave;
    if (edge >= ET) return;
    const int src = (edge < E_EDGES) ? ei[edge] : edge - E_EDGES;
    const int dst = (edge < E_EDGES) ? ei[E_EDGES + edge] : edge - E_EDGES;
    const float* pl = xl + (size_t)src * (4 * H);
    const float* pr = xr + (size_t)dst * (4 * H);
#pragma unroll
    for (int h = 0; h < 4; ++h) {
        float p = 0.f;
#pragma unroll
        for (int j = 0; j < 4; ++j) {
            const int c = h * H + lane + 32 * j;
            p += att[c] * lrelu(pl[c] + pr[c]);
        }
        p = wave_sum(p);
        if (lane == 0) e1[(size_t)edge * 4 + h] = p;
    }
}

// ---------------- layer-1 segment softmax + aggregation (per dst node) --------
// block = 128 threads = 4 waves, wave w handles head w; lanes own 4 channels each.
__global__ void k_agg1(const int* __restrict__ ei, const int* __restrict__ offs,
                       const int* __restrict__ eid, const float* __restrict__ e1,
                       const float* __restrict__ xl, const float* __restrict__ b1,
                       float* __restrict__ x1) {
    const int n = blockIdx.x;
    const int w = threadIdx.x >> 5, lane = threadIdx.x & 31;
    const int beg = offs[n], end = offs[n + 1];   // deg >= 1 (self loop)
    float mx = -3.4e38f;
    for (int i = beg + lane; i < end; i += 32) mx = fmaxf(mx, e1[(size_t)eid[i] * 4 + w]);
    mx = wave_max(mx);
    float s = 0.f;
    for (int i = beg + lane; i < end; i += 32) s += __expf(e1[(size_t)eid[i] * 4 + w] - mx);
    s = wave_sum(s);
    const float inv = 1.f / (s + 1e-16f);
    float a0 = 0.f, a1 = 0.f, a2 = 0.f, a3 = 0.f;
    for (int i = beg; i < end; ++i) {
        const int ed = eid[i];
        const int src = (ed < E_EDGES) ? ei[ed] : ed - E_EDGES;
        const float al = __expf(e1[(size_t)ed * 4 + w] - mx) * inv;
        const float* p = xl + (size_t)src * (4 * H) + w * H;
        a0 += al * p[lane];
        a1 += al * p[lane + 32];
        a2 += al * p[lane + 64];
        a3 += al * p[lane + 96];
    }
    float* o = x1 + (size_t)n * (4 * H) + w * H;
    const float* b = b1 + w * H;
    o[lane]      = fmaxf(a0 + b[lane], 0.f);
    o[lane + 32] = fmaxf(a1 + b[lane + 32], 0.f);
    o[lane + 64] = fmaxf(a2 + b[lane + 64], 0.f);
    o[lane + 96] = fmaxf(a3 + b[lane + 96], 0.f);
}

// ---------------- layer-2 edge scores: e2[E] ----------------------------------
__global__ void k_scores2(const int* __restrict__ ei, const float* __restrict__ xl,
                          const float* __restrict__ xr, const float* __restrict__ att,
                          float* __restrict__ e2) {
    const int wave = threadIdx.x >> 5, lane = threadIdx.x & 31;
    const int edge = blockIdx.x * 8 + wave;
    if (edge >= ET) return;
    const int src = (edge < E_EDGES) ? ei[edge] : edge - E_EDGES;
    const int dst = (edge < E_EDGES) ? ei[E_EDGES + edge] : edge - E_EDGES;
    const float* pl = xl + (size_t)src * H;
    const float* pr = xr + (size_t)dst * H;
    float p = 0.f;
#pragma unroll
    for (int j = 0; j < 4; ++j) {
        const int c = lane + 32 * j;
        p += att[c] * lrelu(pl[c] + pr[c]);
    }
    p = wave_sum(p);
    if (lane == 0) e2[edge] = p;
}

// ---------------- layer-2 softmax + aggregation: one wave per dst node --------
__global__ void k_agg2(const int* __restrict__ ei, const int* __restrict__ offs,
                       const int* __restrict__ eid, const float* __restrict__ e2,
                       const float* __restrict__ xl, const float* __restrict__ b2,
                       float* __restrict__ x2) {
    const int n = blockIdx.x;
    const int lane = threadIdx.x & 31;
    const int beg = offs[n], end = offs[n + 1];
    float mx = -3.4e38f;
    for (int i = beg + lane; i < end; i += 32) mx = fmaxf(mx, e2[eid[i]]);
    mx = wave_max(mx);
    float s = 0.f;
    for (int i = beg + lane; i < end; i += 32) s += __expf(e2[eid[i]] - mx);
    s = wave_sum(s);
    const float inv = 1.f / (s + 1e-16f);
    float a0 = 0.f, a1 = 0.f, a2 = 0.f, a3 = 0.f;
    for (int i = beg; i < end; ++i) {
        const int ed = eid[i];
        const int src = (ed < E_EDGES) ? ei[ed] : ed - E_EDGES;
        const float al = __expf(e2[ed] - mx) * inv;
        const float* p = xl + (size_t)src * H;
        a0 += al * p[lane];
        a1 += al * p[lane + 32];
        a2 += al * p[lane + 64];
        a3 += al * p[lane + 96];
    }
    float* o = x2 + (size_t)n * H;
    o[lane]      = a0 + b2[lane];
    o[lane + 32] = a1 + b2[lane + 32];
    o[lane + 64] = a2 + b2[lane + 64];
    o[lane + 96] = a3 + b2[lane + 96];
}

// =============================================================================
extern "C" void kernel_launch(void* const* d_in, const int* in_sizes, int n_in,
                              void* d_out, int out_size, void* d_ws, size_t ws_size,
                              hipStream_t stream) {
    (void)in_sizes; (void)n_in; (void)out_size; (void)ws_size;
    const float* xp    = (const float*)d_in[0];   // [N, 65]
    const int*   ei    = (const int*)d_in[1];     // [2, E]
    const float* emb   = (const float*)d_in[2];   // [5001, 128]
    const float* projW = (const float*)d_in[3];   // [192, 128]
    const float* projb = (const float*)d_in[4];   // [128]
    const float* W1l   = (const float*)d_in[5];   // [128, 512]
    const float* W1r   = (const float*)d_in[6];   // [128, 512]
    const float* att1  = (const float*)d_in[7];   // [4, 128]
    const float* b1    = (const float*)d_in[8];   // [512]
    const float* W2l   = (const float*)d_in[9];   // [512, 128]
    const float* W2r   = (const float*)d_in[10];  // [512, 128]
    const float* att2  = (const float*)d_in[11];  // [1, 128]
    const float* b2    = (const float*)d_in[12];  // [128]
    const float* outW  = (const float*)d_in[13];  // [128, 10]
    const float* outb  = (const float*)d_in[14];  // [10]
    float* out = (float*)d_out;                   // [N, 10]

    // ---- carve workspace ----
    char* p = (char*)d_ws;
    auto carveF = [&](size_t n) { float* r = (float*)p; p += n * sizeof(float); return r; };
    auto carveI = [&](size_t n) { int* r = (int*)p; p += n * sizeof(int); return r; };
    float* x    = carveF((size_t)N_NODES * H);        // proj output
    float* xl1  = carveF((size_t)N_NODES * 4 * H);
    float* xr1  = carveF((size_t)N_NODES * 4 * H);
    float* x1   = carveF((size_t)N_NODES * 4 * H);    // layer-1 output
    float* xl2  = carveF((size_t)N_NODES * H);
    float* xr2  = carveF((size_t)N_NODES * H);
    float* x2   = carveF((size_t)N_NODES * H);        // layer-2 output
    float* e1   = carveF((size_t)ET * 4);
    float* e2   = carveF((size_t)ET);
    int* counts = carveI(N_NODES);
    int* offs   = carveI(N_NODES + 1);
    int* cursor = carveI(N_NODES);
    int* eid    = carveI(ET);

    const int MT = N_NODES / 16;  // 1250 row tiles (N divisible by 16)

    // CSR build (independent of GEMMs; issue first)
    k_zero<<<(N_NODES + 255) / 256, 256, 0, stream>>>(counts);
    k_count<<<(ET + 255) / 256, 256, 0, stream>>>(ei, counts);
    k_scan<<<1, 1024, 0, stream>>>(counts, offs, cursor);
    k_scatter<<<(ET + 255) / 256, 256, 0, stream>>>(ei, cursor, eid);

    // Stage 0: gather + projection (K=192, 8 col tiles)
    k_proj<<<dim3(MT, 2), 128, 0, stream>>>(xp, emb, projW, projb, x);

    // Stage 1: xl1/xr1 = x @ W1{l,r}   (K=128, 512 cols -> 32 col tiles)
    k_gemm<<<dim3(MT, 8), 128, 16 * 128 * sizeof(float), stream>>>(
        x, H, W1l, 4 * H, nullptr, xl1, 4 * H, H, 0);
    k_gemm<<<dim3(MT, 8), 128, 16 * 128 * sizeof(float), stream>>>(
        x, H, W1r, 4 * H, nullptr, xr1, 4 * H, H, 0);

    // Layer-1 attention: scores, then per-dst softmax + aggregate (+b1, ReLU)
    k_scores1<<<(ET + 7) / 8, 256, 0, stream>>>(ei, xl1, xr1, att1, e1);
    k_agg1<<<N_NODES, 128, 0, stream>>>(ei, offs, eid, e1, xl1, b1, x1);

    // Stage 2: xl2/xr2 = x1 @ W2{l,r}  (K=512, 128 cols -> 8 col tiles)
    k_gemm<<<dim3(MT, 2), 128, 16 * 512 * sizeof(float), stream>>>(
        x1, 4 * H, W2l, H, nullptr, xl2, H, 4 * H, 0);
    k_gemm<<<dim3(MT, 2), 128, 16 * 512 * sizeof(float), stream>>>(
        x1, 4 * H, W2r, H, nullptr, xr2, H, 4 * H, 0);

    // Layer-2 attention
    k_scores2<<<(ET + 7) / 8, 256, 0, stream>>>(ei, xl2, xr2, att2, e2);
    k_agg2<<<N_NODES, 32, 0, stream>>>(ei, offs, eid, e2, xl2, b2, x2);

    // Output head: out = x2 @ out_W + out_b  (padded-LDS B, fully unrolled)
    k_out<<<MT, 128, 0, stream>>>(x2, outW, outb, out);
}